// AgentAttention_40415642255963
// MI455X (gfx1250) — compile-verified
//
#include <hip/hip_runtime.h>

typedef __attribute__((ext_vector_type(2))) float v2f;
typedef __attribute__((ext_vector_type(8))) float v8f;

#define WMMA_F32(a, b, c) \
  __builtin_amdgcn_wmma_f32_16x16x4_f32(false, (a), false, (b), (short)0, (c), false, false)

static constexpr int BATCH = 8;
static constexpr int SEQ   = 4097;   // n
static constexpr int CH    = 768;    // c
static constexpr int NH    = 12;     // heads
static constexpr int HD    = 64;     // head dim
static constexpr int AG    = 49;     // agents
static constexpr int POOL  = 4096;   // n-1
static constexpr int SEQP  = 4112;   // seq padded to 16
static constexpr int MROWS = BATCH * SEQ;       // 32776
static constexpr int MT    = (MROWS + 15) / 16; // 2049
static constexpr float SCALE = 0.125f;          // 64^-0.5

// 8B-aligned vector load (caller guarantees pointer validity & alignment)
__device__ __forceinline__ v2f ld2(const float* p) {
  return *reinterpret_cast<const v2f*>(p);
}
__device__ __forceinline__ v2f sel2(bool ok, v2f a) {
  v2f z = {0.f, 0.f};
  return ok ? a : z;
}

// ---------------------------------------------------------------------------
// 1) qkv = x @ Wqkv  (M x 2304 = (M x 768) @ (768 x 2304)), split to q/k/v
//    wave tile: 16(M) x 64(N), f32 WMMA 16x16x4, K loop over 768
// ---------------------------------------------------------------------------
__global__ __launch_bounds__(32)
void k_qkv_gemm(const float* __restrict__ x, const float* __restrict__ w,
                float* __restrict__ q, float* __restrict__ kbuf,
                float* __restrict__ v)
{
  const int lane = threadIdx.x;
  const int half = lane >> 4, lm = lane & 15;
  const int col0 = blockIdx.x * 64;   // N blocks fastest: A tile stays hot
  const int row0 = blockIdx.y * 16;
  const int rowA = row0 + lm;
  const bool rok = rowA < MROWS;
  const float* arow = x + (size_t)(rok ? rowA : 0) * CH;

  v8f acc[4] = {v8f{}, v8f{}, v8f{}, v8f{}};
  for (int k0 = 0; k0 < CH; k0 += 4) {
    const int kk = k0 + 2 * half;
    const v2f a = sel2(rok, ld2(arow + kk));           // global_load_b64 + cndmask
    const float* b0 = w + (size_t)kk * (3 * CH) + col0 + lm;
#pragma unroll
    for (int nt = 0; nt < 4; ++nt) {
      v2f bf;
      bf.x = b0[nt * 16];
      bf.y = b0[nt * 16 + 3 * CH];
      acc[nt] = WMMA_F32(a, bf, acc[nt]);
    }
  }
  float* outs[3] = {q, kbuf, v};
#pragma unroll
  for (int nt = 0; nt < 4; ++nt) {
    const int col  = col0 + nt * 16 + lm;
    const int part = col / CH;            // uniform within a 16-col subtile
    const int lcol = col - part * CH;
    float* dst = outs[part];
#pragma unroll
    for (int j = 0; j < 8; ++j) {
      const int r = row0 + 8 * half + j;
      if (r < MROWS) dst[(size_t)r * CH + lcol] = acc[nt][j];
    }
  }
}

// ---------------------------------------------------------------------------
// 2) agent_tokens[b][i][ch] = mean over pooled rows of q[:, :-1, :]
// ---------------------------------------------------------------------------
__global__ __launch_bounds__(256)
void k_pool(const float* __restrict__ q, float* __restrict__ ag)
{
  const int bi = blockIdx.x / AG;
  const int ai = blockIdx.x % AG;
  const int s = ai * POOL / AG;
  const int e = ((ai + 1) * POOL + AG - 1) / AG;
  const float inv = 1.f / (float)(e - s);
  for (int ch = threadIdx.x; ch < CH; ch += blockDim.x) {
    float acc = 0.f;
    for (int r = s; r < e; ++r)
      acc += q[((size_t)bi * SEQ + r) * CH + ch];
    ag[((size_t)bi * AG + ai) * CH + ch] = acc * inv;
  }
}

// ---------------------------------------------------------------------------
// 3) S[bh][i][j] = scale * dot(agent_i, key_j)   (49 x 4097, padded 64 x 4112)
// ---------------------------------------------------------------------------
__global__ __launch_bounds__(32)
void k_agent_scores(const float* __restrict__ ag, const float* __restrict__ kbuf,
                    float* __restrict__ S)
{
  const int lane = threadIdx.x, half = lane >> 4, lm = lane & 15;
  const int bh = blockIdx.z, bi = bh / NH, h = bh % NH;
  const int i0 = blockIdx.y * 16;   // agent tile
  const int j0 = blockIdx.x * 16;   // key tile
  const int ia = i0 + lm;  const bool aok = ia < AG;
  const int jb = j0 + lm;  const bool jok = jb < SEQ;
  const float* arow = ag   + ((size_t)bi * AG  + (aok ? ia : 0)) * CH + h * HD;
  const float* krow = kbuf + ((size_t)bi * SEQ + (jok ? jb : 0)) * CH + h * HD;

  v8f acc = {};
  for (int k0 = 0; k0 < HD; k0 += 4) {
    const int kk = k0 + 2 * half;
    const v2f a  = sel2(aok, ld2(arow + kk));
    const v2f bf = sel2(jok, ld2(krow + kk));
    acc = WMMA_F32(a, bf, acc);
  }
#pragma unroll
  for (int j = 0; j < 8; ++j) {
    const int i = i0 + 8 * half + j;
    if (i < AG) S[((size_t)bh * AG + i) * SEQP + j0 + lm] = acc[j] * SCALE;
  }
}

// ---------------------------------------------------------------------------
// 4) row softmax over 4097 valid keys; pads [4097,4112) set to 0
// ---------------------------------------------------------------------------
__global__ __launch_bounds__(256)
void k_softmax(float* __restrict__ S)
{
  float* s = S + (size_t)blockIdx.x * SEQP;
  const int tid = threadIdx.x;
  __shared__ float red[8];
  __shared__ float bm, bs;

  float m = -1e30f;
  for (int j = tid; j < SEQ; j += 256) m = fmaxf(m, s[j]);
  for (int off = 16; off > 0; off >>= 1) m = fmaxf(m, __shfl_xor(m, off, 32));
  if ((tid & 31) == 0) red[tid >> 5] = m;
  __syncthreads();
  if (tid == 0) { float t = red[0]; for (int i = 1; i < 8; ++i) t = fmaxf(t, red[i]); bm = t; }
  __syncthreads();
  m = bm;

  float sum = 0.f;
  for (int j = tid; j < SEQ; j += 256) sum += __expf(s[j] - m);
  for (int off = 16; off > 0; off >>= 1) sum += __shfl_xor(sum, off, 32);
  if ((tid & 31) == 0) red[tid >> 5] = sum;
  __syncthreads();
  if (tid == 0) { float t = 0.f; for (int i = 0; i < 8; ++i) t += red[i]; bs = t; }
  __syncthreads();
  const float inv = 1.f / bs;

  for (int j = tid; j < SEQP; j += 256)
    s[j] = (j < SEQ) ? __expf(s[j] - m) * inv : 0.f;
}

// ---------------------------------------------------------------------------
// 5) agent_v[bh][i][d] = P(49 x 4112) @ V_h(4112 x 64)
// ---------------------------------------------------------------------------
__global__ __launch_bounds__(32)
void k_agent_v(const float* __restrict__ S, const float* __restrict__ v,
               float* __restrict__ av)
{
  const int lane = threadIdx.x, half = lane >> 4, lm = lane & 15;
  const int bh = blockIdx.z, bi = bh / NH, h = bh % NH;
  const int i0 = blockIdx.y * 16;   // agent tile
  const int d0 = blockIdx.x * 16;   // head-dim tile
  const int ia = i0 + lm;  const bool aok = ia < AG;
  const float* prow = S + ((size_t)bh * AG + (aok ? ia : 0)) * SEQP;
  const float* vh = v + (size_t)bi * SEQ * CH + h * HD + d0 + lm;

  v8f acc = {};
  for (int k0 = 0; k0 < SEQP; k0 += 4) {
    const int kk = k0 + 2 * half;
    const v2f a = sel2(aok, ld2(prow + kk));           // P pads are exact zeros
    const int j0r = (kk     < SEQ) ? kk     : 0;
    const int j1r = (kk + 1 < SEQ) ? kk + 1 : 0;
    const float t0 = vh[(size_t)j0r * CH];
    const float t1 = vh[(size_t)j1r * CH];
    v2f bf;
    bf.x = (kk     < SEQ) ? t0 : 0.f;
    bf.y = (kk + 1 < SEQ) ? t1 : 0.f;
    acc = WMMA_F32(a, bf, acc);
  }
#pragma unroll
  for (int j = 0; j < 8; ++j) {
    const int i = i0 + 8 * half + j;
    if (i < AG) av[((size_t)bh * AG + i) * HD + d0 + lm] = acc[j];
  }
}

// ---------------------------------------------------------------------------
// 6) fused: scores = Q_h(16 x 64) @ Ag_h^T (64 agents padded) -> softmax ->
//    out = P(16 x 64) @ agent_v(64 x 64), one 16-row tile per wave
// ---------------------------------------------------------------------------
__global__ __launch_bounds__(32)
void k_qattn_out(const float* __restrict__ q, const float* __restrict__ ag,
                 const float* __restrict__ av, float* __restrict__ out)
{
  __shared__ float pbuf[16 * 66];   // stride 66: 8B aligned + conflict-free
  const int lane = threadIdx.x, half = lane >> 4, lm = lane & 15;
  const int bh = blockIdx.y, bi = bh / NH, h = bh % NH;
  const int r0 = blockIdx.x * 16;
  const int r = r0 + lm;  const bool rok = r < SEQ;
  const float* qrow = q + ((size_t)bi * SEQ + (rok ? r : 0)) * CH + h * HD;

  v8f sc[4] = {v8f{}, v8f{}, v8f{}, v8f{}};
  for (int k0 = 0; k0 < HD; k0 += 4) {
    const int kk = k0 + 2 * half;
    const v2f a = sel2(rok, ld2(qrow + kk));
#pragma unroll
    for (int nt = 0; nt < 4; ++nt) {
      const int j = nt * 16 + lm;            // agent index (col)
      const bool jok = j < AG;
      const float* agrow = ag + ((size_t)bi * AG + (jok ? j : 0)) * CH + h * HD;
      const v2f bf = sel2(jok, ld2(agrow + kk));
      sc[nt] = WMMA_F32(a, bf, sc[nt]);
    }
  }

  // per-row softmax over 64 padded agent columns (cols >= 49 masked out)
#pragma unroll
  for (int jv = 0; jv < 8; ++jv) {
    float m = -1e30f;
#pragma unroll
    for (int nt = 0; nt < 4; ++nt) {
      float val = sc[nt][jv] * SCALE;
      if (nt * 16 + lm >= AG) val = -1e30f;
      sc[nt][jv] = val;
      m = fmaxf(m, val);
    }
#pragma unroll
    for (int off = 8; off > 0; off >>= 1) m = fmaxf(m, __shfl_xor(m, off, 32));
    float ssum = 0.f;
#pragma unroll
    for (int nt = 0; nt < 4; ++nt) {
      const float p = __expf(sc[nt][jv] - m);
      sc[nt][jv] = p;
      ssum += p;
    }
#pragma unroll
    for (int off = 8; off > 0; off >>= 1) ssum += __shfl_xor(ssum, off, 32);
    const float inv = 1.f / ssum;
#pragma unroll
    for (int nt = 0; nt < 4; ++nt)
      pbuf[(8 * half + jv) * 66 + nt * 16 + lm] = sc[nt][jv] * inv;
  }
  __syncthreads();

  // out(16 x 64) = P(16 x 64) @ agent_v(64 x 64, rows >= 49 are zero)
  v8f o[4] = {v8f{}, v8f{}, v8f{}, v8f{}};
  for (int k0 = 0; k0 < 64; k0 += 4) {
    const int kk = k0 + 2 * half;              // agent index (K)
    const v2f a = ld2(&pbuf[lm * 66 + kk]);    // ds_load_b64, conflict-free
    const int a0 = (kk     < AG) ? kk     : 0;
    const int a1 = (kk + 1 < AG) ? kk + 1 : 0;
#pragma unroll
    for (int nt = 0; nt < 4; ++nt) {
      const int d = nt * 16 + lm;
      const float t0 = av[((size_t)bh * AG + a0) * HD + d];
      const float t1 = av[((size_t)bh * AG + a1) * HD + d];
      v2f bf;
      bf.x = (kk     < AG) ? t0 : 0.f;
      bf.y = (kk + 1 < AG) ? t1 : 0.f;
      o[nt] = WMMA_F32(a, bf, o[nt]);
    }
  }
#pragma unroll
  for (int nt = 0; nt < 4; ++nt)
#pragma unroll
    for (int j = 0; j < 8; ++j) {
      const int rr = r0 + 8 * half + j;
      if (rr < SEQ)
        out[((size_t)bi * SEQ + rr) * CH + h * HD + nt * 16 + lm] = o[nt][j];
    }
}

// ---------------------------------------------------------------------------
// 7) out[:, :-1, :] += depthwise_conv1d(v[:, :-1, :], k=3, pad=1) + bias
// ---------------------------------------------------------------------------
__global__ __launch_bounds__(256)
void k_dwc_add(const float* __restrict__ v, const float* __restrict__ w,
               const float* __restrict__ bias, float* __restrict__ out)
{
  const size_t idx = (size_t)blockIdx.x * blockDim.x + threadIdx.x;
  const size_t total = (size_t)BATCH * POOL * CH;
  if (idx >= total) return;
  const int ch = (int)(idx % CH);
  const int t  = (int)((idx / CH) % POOL);
  const int bi = (int)(idx / ((size_t)CH * POOL));
  float acc = bias[ch];
#pragma unroll
  for (int kk = 0; kk < 3; ++kk) {
    const int tt = t - 1 + kk;
    if (tt >= 0 && tt < POOL)
      acc += w[ch * 3 + kk] * v[((size_t)bi * SEQ + tt) * CH + ch];
  }
  out[((size_t)bi * SEQ + t) * CH + ch] += acc;
}

// ---------------------------------------------------------------------------
// 8) d_out = attn @ Wproj + bproj   (M x 768 = (M x 768) @ (768 x 768))
// ---------------------------------------------------------------------------
__global__ __launch_bounds__(32)
void k_proj(const float* __restrict__ xin, const float* __restrict__ w,
            const float* __restrict__ bias, float* __restrict__ out)
{
  const int lane = threadIdx.x, half = lane >> 4, lm = lane & 15;
  const int col0 = blockIdx.x * 64;   // N blocks fastest
  const int row0 = blockIdx.y * 16;
  const int rowA = row0 + lm;
  const bool rok = rowA < MROWS;
  const float* arow = xin + (size_t)(rok ? rowA : 0) * CH;

  v8f acc[4] = {v8f{}, v8f{}, v8f{}, v8f{}};
  for (int k0 = 0; k0 < CH; k0 += 4) {
    const int kk = k0 + 2 * half;
    const v2f a = sel2(rok, ld2(arow + kk));
    const float* b0 = w + (size_t)kk * CH + col0 + lm;
#pragma unroll
    for (int nt = 0; nt < 4; ++nt) {
      v2f bf;
      bf.x = b0[nt * 16];
      bf.y = b0[nt * 16 + CH];
      acc[nt] = WMMA_F32(a, bf, acc[nt]);
    }
  }
#pragma unroll
  for (int nt = 0; nt < 4; ++nt) {
    const int col = col0 + nt * 16 + lm;
    const float bv = bias[col];
#pragma unroll
    for (int j = 0; j < 8; ++j) {
      const int r = row0 + 8 * half + j;
      if (r < MROWS) out[(size_t)r * CH + col] = acc[nt][j] + bv;
    }
  }
}

// ---------------------------------------------------------------------------
extern "C" void kernel_launch(void* const* d_in, const int* in_sizes, int n_in,
                              void* d_out, int out_size, void* d_ws, size_t ws_size,
                              hipStream_t stream)
{
  (void)in_sizes; (void)n_in; (void)out_size; (void)ws_size;
  const float* x     = (const float*)d_in[0];
  const float* Wqkv  = (const float*)d_in[1];
  const float* Wproj = (const float*)d_in[2];
  const float* bproj = (const float*)d_in[3];
  const float* dwcw  = (const float*)d_in[4];
  const float* dwcb  = (const float*)d_in[5];
  float* out = (float*)d_out;
  float* ws  = (float*)d_ws;

  const size_t NE = (size_t)BATCH * SEQ * CH;      // 25,171,968 floats
  float* q    = ws;
  float* kbuf = ws + NE;
  float* v    = ws + 2 * NE;
  float* attn = ws + 3 * NE;                       // pre-proj buffer
  float* S    = attn;                              // aliased: S consumed before attn is written
  float* ag   = ws + 4 * NE;                       // b*a*c   = 301,056
  float* av   = ag + (size_t)BATCH * AG * CH;      // b*h*a*hd= 301,056

  dim3 g1((3 * CH) / 64, MT);
  k_qkv_gemm<<<g1, 32, 0, stream>>>(x, Wqkv, q, kbuf, v);

  k_pool<<<BATCH * AG, 256, 0, stream>>>(q, ag);

  dim3 g3(SEQP / 16, 4, BATCH * NH);
  k_agent_scores<<<g3, 32, 0, stream>>>(ag, kbuf, S);

  k_softmax<<<BATCH * NH * AG, 256, 0, stream>>>(S);

  dim3 g5(HD / 16, 4, BATCH * NH);
  k_agent_v<<<g5, 32, 0, stream>>>(S, v, av);

  dim3 g6((SEQ + 15) / 16, BATCH * NH);
  k_qattn_out<<<g6, 32, 0, stream>>>(q, ag, av, attn);

  const size_t tot = (size_t)BATCH * POOL * CH;
  k_dwc_add<<<(unsigned)((tot + 255) / 256), 256, 0, stream>>>(v, dwcw, dwcb, attn);

  dim3 g8(CH / 64, MT);
  k_proj<<<g8, 32, 0, stream>>>(attn, Wproj, bproj, out);
}